// LindbladLoss_51049981280670
// MI455X (gfx1250) — compile-verified
//
#include <hip/hip_runtime.h>

// MI455X (gfx1250, wave32). Bandwidth-bound segmented quaternion product:
//   stage1: 256 blocks x 256 threads, each thread folds 32 steps (24 MB stream,
//           ~1us at 23.3 TB/s), then a WMMA_F32_16X16X4_F32 block-diagonal
//           tree combines 256 chunk products -> 1 per block.
//   stage2: one block combines the 256 block products the same way and emits
//           (total_loss, fidelity).
// Combine law: chunk products are quaternions q; represent as 4x4 real
// left-multiplication matrices L(q) (homomorphism), so ordered pair-combine is
// a 4x4 real matmul -> 4 of them per v_wmma_f32_16x16x4_f32.

typedef float v2f __attribute__((ext_vector_type(2)));
typedef float v8f __attribute__((ext_vector_type(8)));

#define NBLK 256
#define TPB  256

__device__ __forceinline__ void quat_to_L(float w, float x, float y, float z,
                                          float* L /*16 floats, row-major*/) {
  L[0]  = w;  L[1]  = -x; L[2]  = -y; L[3]  = -z;
  L[4]  = x;  L[5]  =  w; L[6]  = -z; L[7]  =  y;
  L[8]  = y;  L[9]  =  z; L[10] =  w; L[11] = -x;
  L[12] = z;  L[13] = -y; L[14] =  x; L[15] =  w;
}

// Ordered pairwise product tree over TPB 4x4 matrices held in LDS.
// Pair p: slot[p] = slot[2p+1] @ slot[2p]   (later-in-time on the left).
// Wave 0 does the math; one WMMA handles 4 pairs (block-diagonal extraction).
// Must be called from a block-uniform point (contains __syncthreads).
__device__ __forceinline__ void wmma_tree_combine(float (*Lm)[16]) {
  int count = TPB;
  while (count > 1) {
    const int npairs = count >> 1;
    if (threadIdx.x < 32) {                 // full wave 0: EXEC all-ones at WMMA
      const int l   = threadIdx.x;
      const int r16 = l & 15;               // A row / B col (0..15)
      const int kb  = (l < 16) ? 0 : 2;     // K pair per documented layout
      const int sub = r16 >> 2;             // which 4x4 block (0..3)
      const int rr  = r16 & 3;              // row/col within block
      for (int p0 = 0; p0 < npairs; p0 += 4) {
        int m = p0 + sub; if (m > npairs - 1) m = npairs - 1;  // clamp tail
        const float* LA = Lm[2 * m + 1];    // later  -> A (16x4, stacked)
        const float* LB = Lm[2 * m];        // earlier-> B (4x16, side-by-side)
        v2f a, b;
        a[0] = LA[rr * 4 + kb];     a[1] = LA[rr * 4 + kb + 1];
        b[0] = LB[kb * 4 + rr];     b[1] = LB[(kb + 1) * 4 + rr];
        v8f c = {};
        v8f d = __builtin_amdgcn_wmma_f32_16x16x4_f32(
            false, a, false, b, (short)0, c, false, false);
        // D layout: VGPR v -> M = v (lanes 0-15) or v+8 (lanes 16-31), N = l&15.
        // Diagonal blocks live in lanes 0-7 (blocks 0,1) and 24-31 (blocks 2,3).
        const bool active = (l < 8) || (l >= 24);
        const int  blk = r16 >> 2;                 // destination pair in group
        const int  hi  = (l >> 2) & 1;             // rows 0-3 vs 4-7 of VGPRs
        if (active && (p0 + blk) < npairs) {
          float* LD = Lm[p0 + blk];
          const int col = r16 & 3;
          LD[0 * 4 + col] = hi ? d[4] : d[0];
          LD[1 * 4 + col] = hi ? d[5] : d[1];
          LD[2 * 4 + col] = hi ? d[6] : d[2];
          LD[3 * 4 + col] = hi ? d[7] : d[3];
        }
      }
    }
    __syncthreads();
    count = npairs;
  }
}

__global__ void __launch_bounds__(TPB)
lind_stage1(const float* __restrict__ pulses, const float* __restrict__ dtp,
            int n, int steps_per_thread, float* __restrict__ ws) {
  __shared__ float Lm[TPB][16];
  __shared__ float rg[TPB];
  const int tid = threadIdx.x;
  const long gt = (long)blockIdx.x * TPB + tid;

  const float dt    = dtp[0];
  const float p     = 0.01f * dt * 1e9f;   // DEPOL_RATE * dt * 1e9
  const float s     = 1.0f - p;
  const float sdt   = s * dt;
  const float sdt_h = 0.5f * sdt;

  long start = gt * steps_per_thread;
  long end   = start + steps_per_thread;
  if (start > n) start = n;
  if (end   > n) end   = n;

  float qw = 1.f, qx = 0.f, qy = 0.f, qz = 0.f;
  float reg = 0.f, pvx = 0.f, pvy = 0.f, pvz = 0.f;
  const float* pp = pulses + start * 3;

  __builtin_prefetch(pp, 0, 0);                       // global_prefetch_b8
  __builtin_prefetch((const char*)pp + 128, 0, 0);    // chunk spans 3 lines
  __builtin_prefetch((const char*)pp + 256, 0, 0);

  for (long g = start; g < end; ++g) {
    const float ox = pp[0], oy = pp[1], dz = pp[2];
    pp += 3;
    if (g > start) {
      const float dx = ox - pvx, dy = oy - pvy, dd = dz - pvz;
      reg += dx * dx + dy * dy + dd * dd;
    }
    pvx = ox; pvy = oy; pvz = dz;
    // step quaternion m = (s, s*dt*Ox, s*dt*Oy, s*dt*0.5*Delta); q <- m * q
    const float mw = s, mx = sdt * ox, my = sdt * oy, mz = sdt_h * dz;
    const float nw = mw * qw - mx * qx - my * qy - mz * qz;
    const float nx = mw * qx + mx * qw + my * qz - mz * qy;
    const float ny = mw * qy - mx * qz + my * qw + mz * qx;
    const float nz = mw * qz + mx * qy - my * qx + mz * qw;
    qw = nw; qx = nx; qy = ny; qz = nz;
  }
  if (end < n && end > start) {                       // boundary diff at end-1
    const float dx = pp[0] - pvx, dy = pp[1] - pvy, dd = pp[2] - pvz;
    reg += dx * dx + dy * dy + dd * dd;
  }

  quat_to_L(qw, qx, qy, qz, Lm[tid]);
  rg[tid] = reg;
  __syncthreads();

  for (int st = TPB / 2; st > 0; st >>= 1) {          // deterministic reg tree
    if (tid < st) rg[tid] += rg[tid + st];
    __syncthreads();
  }
  wmma_tree_combine(Lm);

  if (tid == 0) {
    float* w = ws + (size_t)blockIdx.x * 8;
    w[0] = Lm[0][0]; w[1] = Lm[0][4]; w[2] = Lm[0][8]; w[3] = Lm[0][12]; // q
    w[4] = rg[0];
  }
}

__global__ void __launch_bounds__(TPB)
lind_stage2(const float* __restrict__ ws, int n, float* __restrict__ out) {
  __shared__ float Lm[TPB][16];
  __shared__ float rg[TPB];
  const int tid = threadIdx.x;
  const float* w = ws + (size_t)tid * 8;
  quat_to_L(w[0], w[1], w[2], w[3], Lm[tid]);
  rg[tid] = w[4];
  __syncthreads();

  for (int st = TPB / 2; st > 0; st >>= 1) {
    if (tid < st) rg[tid] += rg[tid + st];
    __syncthreads();
  }
  wmma_tree_combine(Lm);

  if (tid == 0) {
    const float W = Lm[0][0], Z = Lm[0][12];          // q_total = col 0 of L
    const float fid = 0.25f * (W * W + Z * Z);        // |2*V00|^2 / 16
    const float reg_mean = rg[0] / ((float)(n - 1) * 3.0f);
    out[0] = 1.0f - fid + 0.01f * reg_mean;           // total_loss
    out[1] = fid;                                     // fidelity
  }
}

extern "C" void kernel_launch(void* const* d_in, const int* in_sizes, int n_in,
                              void* d_out, int out_size, void* d_ws, size_t ws_size,
                              hipStream_t stream) {
  const float* pulses = (const float*)d_in[0];
  const float* dtp    = (const float*)d_in[1];
  const int n = in_sizes[0] / 3;                      // 2,097,152 steps
  const int total_threads = NBLK * TPB;               // 65536
  const int K = (n + total_threads - 1) / total_threads;  // 32
  float* ws = (float*)d_ws;                           // NBLK*8 floats = 8 KB
  lind_stage1<<<NBLK, TPB, 0, stream>>>(pulses, dtp, n, K, ws);
  lind_stage2<<<1, TPB, 0, stream>>>(ws, n, (float*)d_out);
}